// PlenoxelModel_84705345012266
// MI455X (gfx1250) — compile-verified
//
#include <hip/hip_runtime.h>
#include <math.h>

// ---------------------------------------------------------------------------
// Plenoxel volume rendering for MI455X (gfx1250, wave32).
//
// Phase 1 (gather-bound, dominant): one workgroup per ray (128 threads = 128
//   samples). Each thread does the 8-corner x 28-channel trilinear gather
//   (7x b128 loads per corner), computes att = exp(-sigma*dt), a block-wide
//   inclusive scan for transmittance (wave32 shfl scan + LDS wave prefix),
//   then folds weight into the 27 SH-coefficient channels and tree-reduces
//   to a per-ray 27-vector W[r][27] in d_ws.
//
// Phase 2 (WMMA): out[16384x3] = W[16384x27] x Bsh[27x3] using
//   V_WMMA_F32_16X16X4_F32, one 16-ray tile per wave, as a chain of 27
//   rank-1 (K=0-only) accumulating WMMAs so only the documented A/C/D
//   layouts and the unambiguous K=0 row of B are relied upon. All loads are
//   unconditional (clamped lane index) + branchless select so no exec
//   save/restore sequences interleave the WMMA chain.
// ---------------------------------------------------------------------------

#define PLX_GX 128
#define PLX_GY 128
#define PLX_GZ 128
#define PLX_VD 28
#define PLX_R  16384
#define PLX_S  128

typedef __attribute__((ext_vector_type(2))) float v2f;
typedef __attribute__((ext_vector_type(8))) float v8f;

__device__ __forceinline__ void plx_sh_basis(float theta, float phi, float sh[9]) {
    const float st = sinf(theta), ct = cosf(theta);
    const float sp = sinf(phi),   cp = cosf(phi);
    sh[0] = 0.28209479177387814f;                         // Y00
    sh[1] = 0.4886025119029199f * st * sp;                // Y_m1_1
    sh[2] = 0.4886025119029199f * ct;                     // Y_0_1
    sh[3] = 0.4886025119029199f * st * cp;                // Y_1_1
    sh[4] = 1.0925484305920792f * st * cp * st * sp;      // Y_m2_2
    sh[5] = 1.0925484305920792f * st * sp * ct;           // Y_m1_2
    sh[6] = 0.31539156525252005f * (3.f * ct * ct - 1.f); // Y_0_2
    sh[7] = 1.0925484305920792f * st * cp * ct;           // Y_1_2
    sh[8] = 0.5462742152960396f * ((st * cp) * (st * cp) - (st * sp) * (st * sp)); // Y_2_2
}

// ---------------------------------------------------------------------------
// Phase 1: per-ray rendering. Block = 1 ray, thread = 1 sample.
// Writes W[r*27 + j] = sum_s weight[r,s] * coeff_j(interp at sample s).
// ---------------------------------------------------------------------------
__global__ __launch_bounds__(PLX_S) void plx_render(
    const float* __restrict__ grid,     // [128,128,128,28]
    const float* __restrict__ origins,  // [R,3]
    const float* __restrict__ dirs,     // [R,3]
    const float* __restrict__ ts,       // [S]
    float* __restrict__ W)              // [R,27] out
{
    const int r = blockIdx.x;
    const int s = threadIdx.x;

    const float t  = ts[s];
    const float dt = (s + 1 < PLX_S) ? (ts[s + 1] - t) : (t - ts[s - 1]);

    const float ox = origins[3 * r + 0], oy = origins[3 * r + 1], oz = origins[3 * r + 2];
    const float dx = dirs[3 * r + 0],    dy = dirs[3 * r + 1],    dz = dirs[3 * r + 2];

    const float px = ox + t * dx, py = oy + t * dy, pz = oz + t * dz;
    const float gx0 = floorf(px), gy0 = floorf(py), gz0 = floorf(pz);
    const int   ix0 = (int)gx0,   iy0 = (int)gy0,   iz0 = (int)gz0;
    const float fx = px - gx0, fy = py - gy0, fz = pz - gz0;
    const float wxs[2] = {1.f - fx, fx};
    const float wys[2] = {1.f - fy, fy};
    const float wzs[2] = {1.f - fz, fz};

    float acc[PLX_VD];
#pragma unroll
    for (int j = 0; j < PLX_VD; ++j) acc[j] = 0.f;

#pragma unroll
    for (int cx = 0; cx < 2; ++cx)
#pragma unroll
    for (int cy = 0; cy < 2; ++cy)
#pragma unroll
    for (int cz = 0; cz < 2; ++cz) {
        const int ix = ix0 + cx, iy = iy0 + cy, iz = iz0 + cz;
        const bool valid = (ix >= 0) & (ix < PLX_GX) &
                           (iy >= 0) & (iy < PLX_GY) &
                           (iz >= 0) & (iz < PLX_GZ);
        if (valid) {
            const float w = wxs[cx] * wys[cy] * wzs[cz];
            const size_t base = (size_t)(((ix * PLX_GY + iy) * PLX_GZ) + iz) * PLX_VD;
            const float4* g4 = (const float4*)(grid + base);  // 28 floats = 7 x b128
#pragma unroll
            for (int q = 0; q < PLX_VD / 4; ++q) {
                const float4 v = g4[q];
                acc[4 * q + 0] = fmaf(w, v.x, acc[4 * q + 0]);
                acc[4 * q + 1] = fmaf(w, v.y, acc[4 * q + 1]);
                acc[4 * q + 2] = fmaf(w, v.z, acc[4 * q + 2]);
                acc[4 * q + 3] = fmaf(w, v.w, acc[4 * q + 3]);
            }
        }
    }

    const float sigma = acc[0];
    const float att   = expf(-sigma * dt);

    // Inclusive block-wide cumsum of att (wave32 shfl scan + LDS wave prefix).
    float v = att;
#pragma unroll
    for (int off = 1; off < 32; off <<= 1) {
        const float n = __shfl_up(v, off, 32);
        if ((s & 31) >= off) v += n;
    }
    __shared__ float waveSum[PLX_S / 32];
    const int wid = s >> 5;
    if ((s & 31) == 31) waveSum[wid] = v;
    __syncthreads();
    float prefix = 0.f;
    for (int w = 0; w < wid; ++w) prefix += waveSum[w];
    const float T      = v + prefix;           // transmittance[s] (inclusive)
    const float weight = T * (1.f - att);

    // Fold weight into the 27 coefficient channels, tree-reduce over samples.
    __shared__ float red[PLX_S * (PLX_VD - 1)];   // 128*27*4 = 13.5 KB
#pragma unroll
    for (int j = 0; j < PLX_VD - 1; ++j)
        red[s * (PLX_VD - 1) + j] = weight * acc[j + 1];
    __syncthreads();
    for (int off = PLX_S / 2; off >= 1; off >>= 1) {
        if (s < off) {
#pragma unroll
            for (int j = 0; j < PLX_VD - 1; ++j)
                red[s * (PLX_VD - 1) + j] += red[(s + off) * (PLX_VD - 1) + j];
        }
        __syncthreads();
    }
    if (s < PLX_VD - 1)
        W[(size_t)r * (PLX_VD - 1) + s] = red[s];
}

// ---------------------------------------------------------------------------
// Phase 2: out[16384x3] = W[16384x27] x Bsh[27x3] via V_WMMA_F32_16X16X4_F32.
// One wave per 16-ray tile; 27 chained rank-1 WMMAs (only K=0 slot used).
// Loads are unconditional (lane&15 clamp) and masked with a branchless
// select, so the unrolled WMMA chain is free of exec manipulation and the
// 27 loads can all be in flight before/while the chain executes.
// ---------------------------------------------------------------------------
__global__ __launch_bounds__(32) void plx_sh_project(
    const float* __restrict__ W,     // [R,27]
    const float* __restrict__ vang,  // [2] = (theta, phi)
    float* __restrict__ out)         // [R,3]
{
    const int lane    = threadIdx.x;     // 0..31, wave32
    const int baseRay = blockIdx.x * 16;

    float sh[9];
    plx_sh_basis(vang[0], vang[1], sh);

    // Every lane loads a full (clamped) row; lanes 16-31 duplicate rows of
    // lanes 0-15 (same cache lines) and are zeroed by select below.
    const float* row = W + (size_t)(baseRay + (lane & 15)) * 27;
    float wv[27];
#pragma unroll
    for (int k = 0; k < 27; ++k) wv[k] = row[k];
    const float amask = (lane < 16) ? 1.f : 0.f;   // branchless lane mask

    v8f c = {0.f, 0.f, 0.f, 0.f, 0.f, 0.f, 0.f, 0.f};
#pragma unroll
    for (int k = 0; k < 27; ++k) {
        // A (16x4 f32): VGPR0 lanes0-15 = A[M=lane][K=0]; all other K slots 0.
        v2f a; a.x = amask * wv[k]; a.y = 0.f;
        // B (4x16 f32): row K=0 (VGPR0, lanes0-15): Bsh[k][n] = (n==k/9)?sh[k%9]:0.
        const int ch = k / 9, q = k - 9 * ch;
        const float bval = (lane == ch) ? sh[q] : 0.f;   // lanes>=16 never match
        v2f b; b.x = bval; b.y = 0.f;
        // (neg_a, A, neg_b, B, c_mod, C, reuse_a, reuse_b)
        c = __builtin_amdgcn_wmma_f32_16x16x4_f32(false, a, false, b,
                                                  (short)0, c, false, false);
    }

    // C/D layout: lanes0-15: VGPR i = D[M=i][N=lane]; lanes16-31: D[M=8+i][N=lane-16].
    const int N = lane & 15;
    if (N < 3) {
        const int mbase = (lane < 16) ? 0 : 8;
#pragma unroll
        for (int i = 0; i < 8; ++i)
            out[(size_t)(baseRay + mbase + i) * 3 + N] = c[i];
    }
}

// ---------------------------------------------------------------------------
// Inputs (setup_inputs order): voxel_grid f32[128*128*128*28], ray_origins
// f32[16384*3], ray_dirs f32[16384*3], ts f32[128], viewing_angle f32[2].
// Output: f32[16384*3]. Workspace: W = 16384*27 floats (~1.77 MB) in d_ws.
// ---------------------------------------------------------------------------
extern "C" void kernel_launch(void* const* d_in, const int* in_sizes, int n_in,
                              void* d_out, int out_size, void* d_ws, size_t ws_size,
                              hipStream_t stream) {
    const float* grid    = (const float*)d_in[0];
    const float* origins = (const float*)d_in[1];
    const float* dirs    = (const float*)d_in[2];
    const float* ts      = (const float*)d_in[3];
    const float* vang    = (const float*)d_in[4];
    float* out = (float*)d_out;
    float* W   = (float*)d_ws;   // 16384*27 floats, fully written before read

    plx_render<<<PLX_R, PLX_S, 0, stream>>>(grid, origins, dirs, ts, W);
    plx_sh_project<<<PLX_R / 16, 32, 0, stream>>>(W, vang, out);
}